// PeepholeLSTM_46325517254841
// MI455X (gfx1250) — compile-verified
//
#include <hip/hip_runtime.h>
#include <math.h>

// Problem constants (from reference)
#define Bn 64
#define Tn 512
#define In 256
#define Hn 512
#define KTOT 768          // I + H fused K dimension
#define NG 64             // persistent workgroups (one H-slice each)
#define HS 8              // hidden units per workgroup
#define NCOLS 32          // 4 gates * HS columns per workgroup
#define APAD 772          // padded K-stride (+4 skew vs 64 banks, keeps b64 loads aligned)
#define GPAD 36

typedef __attribute__((ext_vector_type(2))) float v2f;
typedef __attribute__((ext_vector_type(8))) float v8f;

// Branch-free gate nonlinearities: v_exp_f32 + v_rcp_f32 (no IEEE div chains,
// no EXEC divergence). rcp(inf)=0 gives exact saturation limits.
__device__ __forceinline__ float sigmoidf_(float x) {
    return __builtin_amdgcn_rcpf(1.0f + __expf(-x));
}
__device__ __forceinline__ float tanhf_(float x) {
    float e = __expf(2.0f * x);
    return 1.0f - 2.0f * __builtin_amdgcn_rcpf(e + 1.0f);
}

// Reset barrier counter + zero h0 ping buffer every call (graph-replay safe).
__global__ void lstm_init_kernel(float* __restrict__ hbuf0, unsigned* __restrict__ counter) {
    int i = blockIdx.x * blockDim.x + threadIdx.x;
    if (i < Bn * Hn) hbuf0[i] = 0.0f;
    if (i == 0) *counter = 0u;
}

__device__ __forceinline__ void stage_x(float* __restrict__ act_lds,
                                        const float* __restrict__ x,
                                        int t, int tid) {
    // x_t -> act_lds[:, 0:In]
    for (int f4 = tid; f4 < Bn * (In / 4); f4 += 256) {
        int b  = f4 / (In / 4);
        int c4 = (f4 % (In / 4)) * 4;
        float4 v = *(const float4*)(x + ((size_t)b * Tn + t) * In + c4);
        *(float4*)(act_lds + b * APAD + c4) = v;
    }
}

__device__ __forceinline__ void stage_h(float* __restrict__ act_lds,
                                        const float* __restrict__ hrd,
                                        int tid) {
    // h -> act_lds[:, In:In+Hn]
    for (int f4 = tid; f4 < Bn * (Hn / 4); f4 += 256) {
        int b  = f4 / (Hn / 4);
        int c4 = (f4 % (Hn / 4)) * 4;
        float4 v = *(const float4*)(hrd + b * Hn + c4);
        *(float4*)(act_lds + b * APAD + In + c4) = v;
    }
}

__launch_bounds__(256, 1)
__global__ void lstm_kernel(const float* __restrict__ x,
                            const float* __restrict__ w_ih,
                            const float* __restrict__ w_hh,
                            const float* __restrict__ w_ci,
                            const float* __restrict__ w_cf,
                            const float* __restrict__ w_co,
                            const float* __restrict__ b_ih,
                            const float* __restrict__ b_hh,
                            float* __restrict__ out,      // [B,T,H] ++ h[B,H] ++ c[B,H]
                            float* __restrict__ hbuf,     // 2 * B*H ping-pong
                            unsigned* __restrict__ counter) {
    extern __shared__ float smem[];
    float* w_lds    = smem;                      // [NCOLS][APAD]  weights, resident all T steps
    float* act_lds  = w_lds + NCOLS * APAD;      // [Bn][APAD]     [x_t | h_{t-1}]
    float* g_lds    = act_lds + Bn * APAD;       // [Bn][GPAD]     gate pre-activations
    float* c_lds    = g_lds + Bn * GPAD;         // [Bn][HS]       cell state (resident)
    float* bias_lds = c_lds + Bn * HS;           // [NCOLS]
    float* wc_lds   = bias_lds + NCOLS;          // [24] = wci|wcf|wco for this slice

    const int tid = threadIdx.x;
    const int wg  = blockIdx.x;       // 0..63
    const int hb  = wg * HS;          // hidden-slice base

    // ---- stage weight slice into LDS once: w_lds[n][k] = W[gcol(n)][k] ----
    for (int f4 = tid; f4 < NCOLS * (KTOT / 4); f4 += 256) {
        int n = f4 / (KTOT / 4);
        int k = (f4 % (KTOT / 4)) * 4;
        int gcol = (n >> 3) * Hn + hb + (n & 7);   // gate q = n>>3, unit j = n&7
        float4 v;
        if (k < In) v = *(const float4*)(w_ih + (size_t)gcol * In + k);
        else        v = *(const float4*)(w_hh + (size_t)gcol * Hn + (k - In));
        *(float4*)(w_lds + n * APAD + k) = v;
    }
    if (tid < NCOLS) {
        int gcol = (tid >> 3) * Hn + hb + (tid & 7);
        bias_lds[tid] = b_ih[gcol] + b_hh[gcol];
    }
    if (tid < HS)        wc_lds[tid] = w_ci[hb + tid];
    else if (tid < 16)   wc_lds[tid] = w_cf[hb + tid - 8];
    else if (tid < 24)   wc_lds[tid] = w_co[hb + tid - 16];
    for (int i = tid; i < Bn * HS; i += 256) c_lds[i] = 0.0f;

    // WMMA operand addressing (f32 16x16x4):
    //  A 16x4: lane<16 -> A[M=lane][k..k+1], lane>=16 -> A[M=lane-16][k+2..k+3]
    //  B 4x16: lane<16 -> B[k..k+1][N=lane], lane>=16 -> B[k+2..k+3][N=lane-16]
    //  (B stored N-major in w_lds so both are one contiguous b64 load)
    const int wave  = tid >> 5;
    const int lane  = tid & 31;
    const int lo    = lane & 15;
    const int kh    = (lane >> 4) << 1;          // 0 or 2
    const int rbase = (wave & 3) * 16;           // M tile
    const int nb    = (wave >> 2) * 16;          // N tile
    const float* aptr = act_lds + (rbase + lo) * APAD + kh;
    const float* bptr = w_lds   + (nb + lo)    * APAD + kh;
    const int m0 = rbase + ((lane >> 4) << 3);   // D row base for this lane half
    const int nn = nb + lo;                      // D column

    const size_t hOff = (size_t)Bn * Tn * Hn;
    const size_t cOff = hOff + (size_t)Bn * Hn;

    // Prologue: stage x_0 and h_0 (hbuf[0] zeroed by init kernel)
    stage_x(act_lds, x, 0, tid);
    stage_h(act_lds, hbuf, tid);

    for (int t = 0; t < Tn; ++t) {
        __syncthreads();   // act_lds fully staged

        // ---- fused GEMM: g = [x_t,h] @ Wslice^T  (exact fp32 on matrix pipes) ----
        v8f acc = {};
        #pragma unroll 8
        for (int k = 0; k < KTOT; k += 4) {
            v2f a  = *(const v2f*)(aptr + k);
            v2f bb = *(const v2f*)(bptr + k);
            acc = __builtin_amdgcn_wmma_f32_16x16x4_f32(
                false, a, false, bb, (short)0, acc, false, false);
        }
        #pragma unroll
        for (int r = 0; r < 8; ++r)
            g_lds[(m0 + r) * GPAD + nn] = acc[r];
        __syncthreads();   // D tiles visible; act_lds reads complete (frees x region)

        // ---- gates + state update (all 4 gates local to this WG) ----
        float* hwr = hbuf + (size_t)((t + 1) & 1) * (Bn * Hn);
        #pragma unroll
        for (int rep = 0; rep < 2; ++rep) {
            int e = tid + rep * 256;      // 512 (b,j) pairs
            int b = e >> 3;
            int j = e & 7;
            float gi = g_lds[b * GPAD + j]      + bias_lds[j];
            float gf = g_lds[b * GPAD + 8 + j]  + bias_lds[8 + j];
            float gg = g_lds[b * GPAD + 16 + j] + bias_lds[16 + j];
            float go = g_lds[b * GPAD + 24 + j] + bias_lds[24 + j];
            float cold = c_lds[b * HS + j];
            float ig = sigmoidf_(gi + wc_lds[j] * cold);
            float fg = sigmoidf_(gf + wc_lds[8 + j] * cold);
            float gv = tanhf_(gg);
            float cn = fg * cold + ig * gv;
            float og = sigmoidf_(go + wc_lds[16 + j] * cn);
            float hn = og * tanhf_(cn);
            c_lds[b * HS + j] = cn;
            hwr[b * Hn + hb + j] = hn;
            out[((size_t)b * Tn + t) * Hn + hb + j] = hn;
            if (t == Tn - 1) {            // fold final h/c emission into last step
                out[hOff + (size_t)b * Hn + hb + j] = hn;
                out[cOff + (size_t)b * Hn + hb + j] = cn;
            }
        }

        if (t + 1 < Tn) {
            // x_{t+1} staging does NOT depend on other WGs -> hoist before barrier
            stage_x(act_lds, x, t + 1, tid);

            // ---- grid-wide barrier (one per step; ping-pong makes it sufficient) ----
            __syncthreads();
            if (tid == 0) {
                __hip_atomic_fetch_add(counter, 1u, __ATOMIC_ACQ_REL, __HIP_MEMORY_SCOPE_AGENT);
                unsigned target = (unsigned)(t + 1) * NG;
                while (__hip_atomic_load(counter, __ATOMIC_ACQUIRE, __HIP_MEMORY_SCOPE_AGENT) < target)
                    __builtin_amdgcn_s_sleep(1);
            }
            __syncthreads();

            // only h staging remains on the post-barrier critical path
            stage_h(act_lds, hbuf + (size_t)((t + 1) & 1) * (Bn * Hn), tid);
        }
    }
}

extern "C" void kernel_launch(void* const* d_in, const int* in_sizes, int n_in,
                              void* d_out, int out_size, void* d_ws, size_t ws_size,
                              hipStream_t stream) {
    (void)in_sizes; (void)n_in; (void)out_size; (void)ws_size;
    const float* x    = (const float*)d_in[0];
    const float* w_ih = (const float*)d_in[1];
    const float* w_hh = (const float*)d_in[2];
    const float* w_ci = (const float*)d_in[3];
    const float* w_cf = (const float*)d_in[4];
    const float* w_co = (const float*)d_in[5];
    const float* b_ih = (const float*)d_in[6];
    const float* b_hh = (const float*)d_in[7];

    float* hbuf = (float*)d_ws;                                  // 2 * B*H floats
    unsigned* counter = (unsigned*)((char*)d_ws + (size_t)2 * Bn * Hn * sizeof(float));

    lstm_init_kernel<<<(Bn * Hn + 255) / 256, 256, 0, stream>>>(hbuf, counter);

    size_t shmem = (size_t)(NCOLS * APAD + Bn * APAD + Bn * GPAD + Bn * HS + NCOLS + 24)
                   * sizeof(float);   // ~308 KB of the 320 KB WGP LDS
    lstm_kernel<<<NG, 256, shmem, stream>>>(x, w_ih, w_hh, w_ci, w_cf, w_co,
                                            b_ih, b_hh, (float*)d_out, hbuf, counter);
}